// FrequencyAwareWaveletTransform_1279900254215
// MI455X (gfx1250) — compile-verified
//
#include <hip/hip_runtime.h>
#include <cstdint>

#define ASYNC_STAGE 1

#if defined(__has_builtin)
#  if __has_builtin(__builtin_amdgcn_wmma_f32_16x16x4_f32)
#    define HAVE_WMMA_F32_16X16X4 1
#  endif
#endif
#ifndef HAVE_WMMA_F32_16X16X4
#  define HAVE_WMMA_F32_16X16X4 0
#endif

typedef __attribute__((ext_vector_type(2))) float v2f;
typedef __attribute__((ext_vector_type(4))) float v4f;
typedef __attribute__((ext_vector_type(8))) float v8f;

// Ext-buffer layout: interior x[i] at e[OFF+i]; left ext e[OFF-6..OFF-1],
// right ext e[OFF+L..OFF+L+6].  OFF=8 makes WMMA result rows 16B-aligned.
#define OFF 8

// db4 rec_lo coefficients (pywt _H); correlation kernels:
//   FLO[k] = H[k]          (low-pass)
//   FHI[k] = (-1)^k H[7-k] (high-pass)
#define H0 (0.23037781330885523f)
#define H1 (0.7148465705525415f)
#define H2 (0.6308807679295904f)
#define H3 (-0.02798376941698385f)
#define H4 (-0.18703481171888114f)
#define H5 (0.030841381835986965f)
#define H6 (0.032883011666982945f)
#define H7 (-0.010597401784997278f)
#define G0 (H7)
#define G1 (-H6)
#define G2 (H5)
#define G3 (-H4)
#define G4 (H3)
#define G5 (-H2)
#define G6 (H1)
#define G7 (-H0)

#define NTHREADS 256
#define NWAVES 8

__device__ __forceinline__ float bsel(int nn, float lo, float hi) {
  return nn == 0 ? lo : (nn == 1 ? hi : 0.0f);
}

// Symmetric (half-sample, drop-first) extension in-place on an ext buffer.
__device__ __forceinline__ void fill_ext(float* e, int L, int t) {
  if (t < 6) {
    e[(OFF - 6) + t] = e[OFF + (5 - t)];   // xp[j] = x[5-j], j=0..5
  } else if (t < 13) {
    const int r = t - 6;
    e[OFF + L + r] = e[OFF + L - 1 - r];   // xp[6+L+r] = x[L-1-r], r=0..6
  }
}

__device__ __forceinline__ void copy_to_global(float* __restrict__ g,
                                               const float* __restrict__ l,
                                               int n, int t) {
  for (int i = t; i < n; i += NTHREADS) g[i] = l[i];
}

// One DWT level: out[f][p] = sum_{k=0..7} F[f][k] * xp[2p + k]
// WMMA mapping: D[16x16] = A[16x4] * B[4x16] + C, twice (k-chunks 0-3, 4-7).
//   A[m,k] = xp[32*tile + 2m + k]   (m = output position within tile)
//   B[k,n] = filter column: n=0 -> FLO, n=1 -> FHI, n>=2 -> 0
// D column n lives in lanes n (rows 0-7) and n+16 (rows 8-15).
// Stores are unguarded: all buffers have tail slack; garbage past Lout is
// either overwritten by fill_ext or never read/copied.
__device__ __forceinline__ void dwt_level(const float* __restrict__ srcExt,
                                          float* __restrict__ dstExt,
                                          float* __restrict__ cdBuf,
                                          int Lout, int lane, int wave,
                                          v2f B0, v2f B1) {
  const int tiles = (Lout + 15) >> 4;
  const int m  = lane & 15;
  const int hi = lane >> 4;
  for (int tile = wave; tile < tiles; tile += NWAVES) {
    const int base = (tile << 5) + (m << 1) + (hi << 1) + (OFF - 6);
#if HAVE_WMMA_F32_16X16X4
    v2f A0, A1;
    A0.x = srcExt[base + 0];  A0.y = srcExt[base + 1];   // k = 2*hi, 2*hi+1
    A1.x = srcExt[base + 4];  A1.y = srcExt[base + 5];   // k = 4+2*hi, ...
    v8f d = {};
    d = __builtin_amdgcn_wmma_f32_16x16x4_f32(false, A0, false, B0, (short)0, d,
                                              false, false);
    d = __builtin_amdgcn_wmma_f32_16x16x4_f32(false, A1, false, B1, (short)0, d,
                                              false, false);
    const int p0 = (tile << 4) + (hi << 3);   // multiple of 8
    const v4f dlo = __builtin_shufflevector(d, d, 0, 1, 2, 3);
    const v4f dhi = __builtin_shufflevector(d, d, 4, 5, 6, 7);
    if (m == 0) {            // column 0 = cA -> ext buffer (16B aligned)
      *(v4f*)(dstExt + OFF + p0)     = dlo;
      *(v4f*)(dstExt + OFF + p0 + 4) = dhi;
    } else if (m == 1) {     // column 1 = cD -> staging buffer (16B aligned)
      *(v4f*)(cdBuf + p0)     = dlo;
      *(v4f*)(cdBuf + p0 + 4) = dhi;
    }
#else
    // VALU fallback: lane (m, hi) computes position tile*16+m for filter hi.
    const int p = (tile << 4) + m;
    const int j = (p << 1) + (OFF - 6);
    if (hi == 0) {
      float acc = H0 * srcExt[j + 0] + H1 * srcExt[j + 1] + H2 * srcExt[j + 2] +
                  H3 * srcExt[j + 3] + H4 * srcExt[j + 4] + H5 * srcExt[j + 5] +
                  H6 * srcExt[j + 6] + H7 * srcExt[j + 7];
      dstExt[OFF + p] = acc;
    } else {
      float acc = G0 * srcExt[j + 0] + G1 * srcExt[j + 1] + G2 * srcExt[j + 2] +
                  G3 * srcExt[j + 3] + G4 * srcExt[j + 4] + G5 * srcExt[j + 5] +
                  G6 * srcExt[j + 6] + G7 * srcExt[j + 7];
      cdBuf[p] = acc;
    }
#endif
  }
}

__global__ __launch_bounds__(NTHREADS) void dwt_db4_kernel(
    const float* __restrict__ x, float* __restrict__ out) {
  constexpr int L0 = 4096;
  constexpr int N1 = 2051, N2 = 1029, N3 = 518, N4 = 262;
  constexpr int NSIG = 64 * 32;
  constexpr size_t BASE_CA4 = 0;
  constexpr size_t BASE_CD4 = BASE_CA4 + (size_t)NSIG * N4;
  constexpr size_t BASE_CD3 = BASE_CD4 + (size_t)NSIG * N4;
  constexpr size_t BASE_CD2 = BASE_CD3 + (size_t)NSIG * N3;
  constexpr size_t BASE_CD1 = BASE_CD2 + (size_t)NSIG * N2;

  __shared__ __align__(16) float bufBig[4144];  // ext(4096)=4109 + slack
  __shared__ __align__(16) float bufMid[2096];  // ext(2051)=2064 + slack
  __shared__ __align__(16) float bufCD[2064];   // cD staging (max 2051)

  const int t    = threadIdx.x;
  const int lane = t & 31;
  const int wave = t >> 5;
  const int s    = blockIdx.x;      // signal id = b*32 + c
  const int b    = s >> 5;
  const int c    = s & 31;
  const float* xin = x + (size_t)b * L0 * 32 + (unsigned)c;

  // Per-lane constant B-matrix columns (col0=FLO, col1=FHI, rest 0).
  // B layout (4x16 f32): VGPR0 = K{0|2}, VGPR1 = K{1|3}; lanes 16-31 -> K+2.
  const int nn = lane & 15;
  const int hh = lane >> 4;
  v2f B0, B1;
  if (hh == 0) {
    B0.x = bsel(nn, H0, G0);  B0.y = bsel(nn, H1, G1);   // chunk0: taps 0,1
    B1.x = bsel(nn, H4, G4);  B1.y = bsel(nn, H5, G5);   // chunk1: taps 4,5
  } else {
    B0.x = bsel(nn, H2, G2);  B0.y = bsel(nn, H3, G3);   // chunk0: taps 2,3
    B1.x = bsel(nn, H6, G6);  B1.y = bsel(nn, H7, G7);   // chunk1: taps 6,7
  }

  // ---- Stage input channel column [L0] (stride 32 floats) into LDS ----
#if ASYNC_STAGE
  {
    uint64_t ga = (uint64_t)(uintptr_t)(xin + (size_t)t * 32);
    uint32_t lp = (uint32_t)(uintptr_t)(&bufBig[OFF + t]);
#pragma unroll
    for (int i = 0; i < L0 / NTHREADS; ++i) {
      asm volatile("global_load_async_to_lds_b32 %0, %1, off" ::"v"(lp), "v"(ga)
                   : "memory");
      ga += (uint64_t)NTHREADS * 32 * 4;   // L stride = 32 floats
      lp += (uint32_t)NTHREADS * 4;
    }
    asm volatile("s_wait_asynccnt 0x0" ::: "memory");
  }
#else
  for (int i = 0; i < L0 / NTHREADS; ++i) {
    const int l = t + i * NTHREADS;
    bufBig[OFF + l] = xin[(size_t)l * 32];
  }
#endif
  __syncthreads();
  fill_ext(bufBig, L0, t);
  __syncthreads();

  // ---- Level 1: bufBig -> cA1 in bufMid, cD1 out ----
  dwt_level(bufBig, bufMid, bufCD, N1, lane, wave, B0, B1);
  __syncthreads();
  copy_to_global(out + BASE_CD1 + (size_t)s * N1, bufCD, N1, t);
  fill_ext(bufMid, N1, t);
  __syncthreads();

  // ---- Level 2: bufMid -> cA2 in bufBig, cD2 out ----
  dwt_level(bufMid, bufBig, bufCD, N2, lane, wave, B0, B1);
  __syncthreads();
  copy_to_global(out + BASE_CD2 + (size_t)s * N2, bufCD, N2, t);
  fill_ext(bufBig, N2, t);
  __syncthreads();

  // ---- Level 3: bufBig -> cA3 in bufMid, cD3 out ----
  dwt_level(bufBig, bufMid, bufCD, N3, lane, wave, B0, B1);
  __syncthreads();
  copy_to_global(out + BASE_CD3 + (size_t)s * N3, bufCD, N3, t);
  fill_ext(bufMid, N3, t);
  __syncthreads();

  // ---- Level 4: bufMid -> cA4 in bufBig, cD4 out ----
  dwt_level(bufMid, bufBig, bufCD, N4, lane, wave, B0, B1);
  __syncthreads();
  copy_to_global(out + BASE_CD4 + (size_t)s * N4, bufCD, N4, t);
  copy_to_global(out + BASE_CA4 + (size_t)s * N4, bufBig + OFF, N4, t);
}

extern "C" void kernel_launch(void* const* d_in, const int* in_sizes, int n_in,
                              void* d_out, int out_size, void* d_ws,
                              size_t ws_size, hipStream_t stream) {
  (void)in_sizes; (void)n_in; (void)d_ws; (void)ws_size; (void)out_size;
  const float* x = (const float*)d_in[0];
  float* out = (float*)d_out;
  dwt_db4_kernel<<<dim3(64 * 32), dim3(NTHREADS), 0, stream>>>(x, out);
}